// JCPOTModel_15006615733810
// MI455X (gfx1250) — compile-verified
//
#include <hip/hip_runtime.h>
#include <math.h>

// JCPOT Sinkhorn OT for MI455X (gfx1250).
//   n_s = n_t = 4096, d = 512, epsilon = 8, 10 Sinkhorn iterations.
// d_out layout: [0] = ot_loss, [1 .. 1+16M) = coupling, [1+16M .. 1+32M) = C.
// d_ws layout (floats): K(16M) | sn(4096) | tn(4096) | u(4096) | v(4096) |
//                       colPartial(32*4096) | lossPartial(16384)

#define NPTS 4096
#define DIM  512
#define NTILES_N32 (NPTS / 32)   // 128 super-tiles of width 32
#define ROWCHUNK 128             // rows per split-K chunk in K^T u
#define NCHUNKS (NPTS / ROWCHUNK) // 32

typedef float v2f __attribute__((ext_vector_type(2)));
typedef float v8f __attribute__((ext_vector_type(8)));

// ---------------------------------------------------------------- row norms
__global__ __launch_bounds__(128)
void jcpot_rownorm(const float* __restrict__ X, float* __restrict__ out) {
    __shared__ float red[128];
    const int row = blockIdx.x;
    const float4* xr = (const float4*)(X + (size_t)row * DIM);
    float4 x = xr[threadIdx.x];                       // 128 float4 per row
    float s = x.x * x.x + x.y * x.y + x.z * x.z + x.w * x.w;
    red[threadIdx.x] = s;
    __syncthreads();
    for (int off = 64; off > 0; off >>= 1) {
        if (threadIdx.x < off) red[threadIdx.x] += red[threadIdx.x + off];
        __syncthreads();
    }
    if (threadIdx.x == 0) out[row] = red[0];
}

// -------------------------------------------- GEMM + cost/kernel epilogue
// One wave computes a 16x32 tile of G = S * T^T with V_WMMA_F32_16X16X4_F32,
// then fuses C = sqrt(max(|s|^2 + |t|^2 - 2g, 0)) and K = exp(-C/8).
__global__ __launch_bounds__(256)
void jcpot_gemm_cost(const float* __restrict__ S, const float* __restrict__ T,
                     const float* __restrict__ sn, const float* __restrict__ tn,
                     float* __restrict__ Cout, float* __restrict__ Kout) {
    const int wave = threadIdx.x >> 5;
    const int lane = threadIdx.x & 31;
    const int tile = blockIdx.x * 8 + wave;          // 32768 wave-tiles
    const int tm  = (tile / NTILES_N32) * 16;
    const int tn0 = (tile % NTILES_N32) * 32;

    // f32 A 16x4 layout: lane<16 holds {K=0,1} of row (lane&15),
    //                    lane>=16 holds {K=2,3}. B mirrors with N.
    const int r    = lane & 15;
    const int koff = (lane >> 4) * 2;

    const float* arow  = S + (size_t)(tm + r)       * DIM + koff;
    const float* brow0 = T + (size_t)(tn0 + r)      * DIM + koff;
    const float* brow1 = T + (size_t)(tn0 + 16 + r) * DIM + koff;

    v8f acc0 = {};
    v8f acc1 = {};
    #pragma unroll 4
    for (int k = 0; k < DIM; k += 4) {
        v2f a  = *(const v2f*)(arow  + k);
        v2f b0 = *(const v2f*)(brow0 + k);
        v2f b1 = *(const v2f*)(brow1 + k);
        acc0 = __builtin_amdgcn_wmma_f32_16x16x4_f32(false, a, false, b0,
                                                     (short)0, acc0, false, false);
        acc1 = __builtin_amdgcn_wmma_f32_16x16x4_f32(false, a, false, b1,
                                                     (short)0, acc1, false, false);
    }

    // D layout: VGPR rr, lanes 0-15 -> M=rr, N=lane; lanes 16-31 -> M=8+rr.
    const int mBase = tm + ((lane >> 4) << 3);
    const int nA    = tn0 + (lane & 15);
    const int nB    = nA + 16;
    const float t2a = tn[nA];
    const float t2b = tn[nB];
    #pragma unroll
    for (int rr = 0; rr < 8; ++rr) {
        const int m = mBase + rr;
        const float s2 = sn[m];
        {
            float g  = acc0[rr];
            float sq = s2 + t2a - 2.0f * g;
            float c  = sqrtf(fmaxf(sq, 0.0f));
            size_t idx = (size_t)m * NPTS + nA;
            Cout[idx] = c;
            Kout[idx] = __expf(-0.125f * c);
        }
        {
            float g  = acc1[rr];
            float sq = s2 + t2b - 2.0f * g;
            float c  = sqrtf(fmaxf(sq, 0.0f));
            size_t idx = (size_t)m * NPTS + nB;
            Cout[idx] = c;
            Kout[idx] = __expf(-0.125f * c);
        }
    }
}

// ---------------------------------------------------------------- Sinkhorn
__global__ __launch_bounds__(256)
void jcpot_v_init(float* __restrict__ v) {
    v[blockIdx.x * 256 + threadIdx.x] = 1.0f;
}

// u_i = (1/n) / (sum_j K_ij v_j + 1e-8): one block per row, float4 streaming.
__global__ __launch_bounds__(256)
void jcpot_u_update(const float* __restrict__ K, const float* __restrict__ v,
                    float* __restrict__ u) {
    __shared__ float red[256];
    const int row = blockIdx.x;
    const float4* kr = (const float4*)(K + (size_t)row * NPTS);
    const float4* v4 = (const float4*)v;
    float s = 0.0f;
    #pragma unroll
    for (int c = 0; c < 4; ++c) {
        const int j = threadIdx.x + c * 256;
        float4 kk = kr[j];
        float4 vv = v4[j];
        s += kk.x * vv.x + kk.y * vv.y + kk.z * vv.z + kk.w * vv.w;
    }
    red[threadIdx.x] = s;
    __syncthreads();
    for (int off = 128; off > 0; off >>= 1) {
        if (threadIdx.x < off) red[threadIdx.x] += red[threadIdx.x + off];
        __syncthreads();
    }
    if (threadIdx.x == 0) u[row] = (1.0f / NPTS) / (red[0] + 1e-8f);
}

// Partial column sums of K^T u: thread-per-column (coalesced), split-K rows.
__global__ __launch_bounds__(256)
void jcpot_v_partial(const float* __restrict__ K, const float* __restrict__ u,
                     float* __restrict__ partial) {
    __shared__ float us[ROWCHUNK];
    const int j  = blockIdx.x * 256 + threadIdx.x;
    const int i0 = blockIdx.y * ROWCHUNK;
    if (threadIdx.x < ROWCHUNK) us[threadIdx.x] = u[i0 + threadIdx.x];
    __syncthreads();
    float s = 0.0f;
    #pragma unroll 4
    for (int i = 0; i < ROWCHUNK; ++i)
        s += K[(size_t)(i0 + i) * NPTS + j] * us[i];
    partial[(size_t)blockIdx.y * NPTS + j] = s;
}

__global__ __launch_bounds__(256)
void jcpot_v_finalize(const float* __restrict__ partial, float* __restrict__ v) {
    const int j = blockIdx.x * 256 + threadIdx.x;
    float s = 0.0f;
    #pragma unroll
    for (int c = 0; c < NCHUNKS; ++c) s += partial[(size_t)c * NPTS + j];
    v[j] = (1.0f / NPTS) / (s + 1e-8f);
}

// ----------------------------------- coupling = u K v^T, fused loss partial
// coup/C live at odd float offsets in d_out -> scalar b32 access for them;
// K (workspace, 16B aligned) is read as float4.
__global__ __launch_bounds__(256)
void jcpot_coupling(const float* __restrict__ K, const float* __restrict__ u,
                    const float* __restrict__ v, const float* __restrict__ Cmat,
                    float* __restrict__ coup, float* __restrict__ lossPartial) {
    __shared__ float red[256];
    const size_t g4  = (size_t)blockIdx.x * 256 + threadIdx.x; // float4 index
    const size_t idx = g4 * 4;
    const int i = (int)(idx >> 12);                 // row (N = 4096 = 2^12)
    float4 k4 = ((const float4*)K)[g4];
    float4 v4 = ((const float4*)v)[g4 & (NPTS / 4 - 1)];
    const float ui = u[i];
    const float kv[4] = {k4.x, k4.y, k4.z, k4.w};
    const float vv[4] = {v4.x, v4.y, v4.z, v4.w};
    float acc = 0.0f;
    #pragma unroll
    for (int e = 0; e < 4; ++e) {
        float cp = ui * kv[e] * vv[e];
        coup[idx + e] = cp;
        acc += cp * Cmat[idx + e];
    }
    red[threadIdx.x] = acc;
    __syncthreads();
    for (int off = 128; off > 0; off >>= 1) {
        if (threadIdx.x < off) red[threadIdx.x] += red[threadIdx.x + off];
        __syncthreads();
    }
    if (threadIdx.x == 0) lossPartial[blockIdx.x] = red[0];
}

__global__ __launch_bounds__(256)
void jcpot_loss_reduce(const float* __restrict__ lossPartial,
                       float* __restrict__ out) {
    __shared__ float red[256];
    float s = 0.0f;
    #pragma unroll
    for (int c = 0; c < 64; ++c) s += lossPartial[threadIdx.x + c * 256];
    red[threadIdx.x] = s;
    __syncthreads();
    for (int off = 128; off > 0; off >>= 1) {
        if (threadIdx.x < off) red[threadIdx.x] += red[threadIdx.x + off];
        __syncthreads();
    }
    if (threadIdx.x == 0)
        out[0] = red[0] * (1.0f / ((float)NPTS * (float)NPTS));
}

// ------------------------------------------------------------------- launch
extern "C" void kernel_launch(void* const* d_in, const int* in_sizes, int n_in,
                              void* d_out, int out_size, void* d_ws, size_t ws_size,
                              hipStream_t stream) {
    (void)in_sizes; (void)n_in; (void)out_size; (void)ws_size;
    const float* S = (const float*)d_in[0];
    const float* T = (const float*)d_in[1];

    float* out   = (float*)d_out;
    float* loss  = out;
    float* coup  = out + 1;
    float* Cmat  = out + 1 + (size_t)NPTS * NPTS;

    float* ws      = (float*)d_ws;
    float* Kmat    = ws;                                  // 16,777,216
    float* sn      = Kmat + (size_t)NPTS * NPTS;          // 4096
    float* tn      = sn + NPTS;                           // 4096
    float* u       = tn + NPTS;                           // 4096
    float* v       = u + NPTS;                            // 4096
    float* partial = v + NPTS;                            // 32 * 4096
    float* lossP   = partial + (size_t)NCHUNKS * NPTS;    // 16384

    jcpot_rownorm<<<NPTS, 128, 0, stream>>>(S, sn);
    jcpot_rownorm<<<NPTS, 128, 0, stream>>>(T, tn);
    jcpot_gemm_cost<<<4096, 256, 0, stream>>>(S, T, sn, tn, Cmat, Kmat);

    jcpot_v_init<<<NPTS / 256, 256, 0, stream>>>(v);
    for (int it = 0; it < 10; ++it) {
        jcpot_u_update<<<NPTS, 256, 0, stream>>>(Kmat, v, u);
        jcpot_v_partial<<<dim3(NPTS / 256, NCHUNKS), 256, 0, stream>>>(Kmat, u, partial);
        jcpot_v_finalize<<<NPTS / 256, 256, 0, stream>>>(partial, v);
    }

    jcpot_coupling<<<(NPTS * (NPTS / 4)) / 256, 256, 0, stream>>>(
        Kmat, u, v, Cmat, coup, lossP);
    jcpot_loss_reduce<<<1, 256, 0, stream>>>(lossP, loss);
}